// ReHubLayer_59923383714405
// MI455X (gfx1250) — compile-verified
//
#include <hip/hip_runtime.h>

// ReHub layer for MI455X (gfx1250, wave32).
// GEMMs use V_WMMA_F32_16X16X4_F32 (fp32 matrix pipe, matches fp32 reference).
// LDS holds W with K-pairs interleaved so each B fragment is a single ds_load_b64.
// Each wave computes 32 rows x 64 cols: 8 WMMAs per (2 global b64 + 4 ds b64).
// Edge aggregation / segment softmax use global float atomics (bandwidth bound).

#define DIM 128
#define CCH 16
#define KCHUNK 128

typedef __attribute__((ext_vector_type(2))) float v2f;
typedef __attribute__((ext_vector_type(8))) float v8f;

// ---------------- utility fills ----------------
__global__ __launch_bounds__(256) void fill_f32(float* __restrict__ p, int n, float v) {
  int i = blockIdx.x * 256 + threadIdx.x;
  if (i < n) p[i] = v;
}
__global__ __launch_bounds__(256) void fill_u32(unsigned* __restrict__ p, int n, unsigned v) {
  int i = blockIdx.x * 256 + threadIdx.x;
  if (i < n) p[i] = v;
}

// ---------------- WMMA fp32 GEMM: OUT = act(IN[nRows,K] @ W[K,Ncols] + bias) ----------------
// block = 256 threads (8 waves); each wave covers 32 rows x 64 cols -> block = 256 rows x 64 cols.
// grid.x = ceil(nRows/256), grid.y = Ncols/64. flags: 1 = relu, 2 = accumulate into OUT.
// LDS: KCHUNK x 64 chunk of W, stored K-pair interleaved:
//   wlds[(k/2)*128 + c*2 + (k&1)] = W[k0+k][colBase+c]
// so the B fragment for lane group (k, k+1) at column c is one aligned 8-byte LDS load.
__global__ __launch_bounds__(256) void gemm_wmma(
    const float* __restrict__ A, const float* __restrict__ W,
    const float* __restrict__ bias, float* __restrict__ out,
    int nRows, int K, int Ncols, int flags) {
  __shared__ float wlds[KCHUNK * 64];
  const int wave   = threadIdx.x >> 5;
  const int lane   = threadIdx.x & 31;
  const int lane15 = lane & 15;
  const int hi     = lane >> 4;           // 0: K pair (k,k+1), 1: K pair (k+2,k+3)
  const int colBase = blockIdx.y * 64;
  const int rowBase = blockIdx.x * 256 + wave * 32;

  int aRow0 = rowBase + lane15;           // clamp loads; EXEC stays all-ones for WMMA
  if (aRow0 >= nRows) aRow0 = nRows - 1;
  int aRow1 = rowBase + 16 + lane15;
  if (aRow1 >= nRows) aRow1 = nRows - 1;

  const v8f vzero = {0.f,0.f,0.f,0.f,0.f,0.f,0.f,0.f};
  v8f acc0[4], acc1[4];
#pragma unroll
  for (int t = 0; t < 4; ++t) { acc0[t] = vzero; acc1[t] = vzero; }

  const float* arow0 = A + (size_t)aRow0 * K;
  const float* arow1 = A + (size_t)aRow1 * K;

  for (int k0 = 0; k0 < K; k0 += KCHUNK) {
    __syncthreads();
    // stage W[k0..k0+128, colBase..colBase+64] into LDS, K-pair interleaved
    for (int i = threadIdx.x; i < KCHUNK * 64; i += 256) {
      int kk = i >> 6, c = i & 63;
      wlds[(kk >> 1) * 128 + c * 2 + (kk & 1)] =
          W[(size_t)(k0 + kk) * Ncols + colBase + c];
    }
    __syncthreads();

    const float* bbase = wlds + (size_t)hi * 128 + lane15 * 2;
#pragma unroll 4
    for (int ks = 0; ks < KCHUNK; ks += 4) {
      // A fragments: lanes 0-15 -> K=(ks,ks+1), lanes 16-31 -> K=(ks+2,ks+3)
      v2f a0 = *(const v2f*)(arow0 + k0 + ks + hi * 2);
      v2f a1 = *(const v2f*)(arow1 + k0 + ks + hi * 2);
      const float* bp = bbase + (ks >> 1) * 128;
#pragma unroll
      for (int t = 0; t < 4; ++t) {
        v2f b = *(const v2f*)(bp + t * 32);   // ds_load_b64, imm offset t*128B
        acc0[t] = __builtin_amdgcn_wmma_f32_16x16x4_f32(
            false, a0, false, b, (short)0, acc0[t], false, false);
        acc1[t] = __builtin_amdgcn_wmma_f32_16x16x4_f32(
            false, a1, false, b, (short)0, acc1[t], false, false);
      }
    }
  }

  // epilogue: C layout -> VGPR r: lanes 0-15 hold M=r, lanes 16-31 hold M=r+8
#pragma unroll
  for (int t = 0; t < 4; ++t) {
    int col = colBase + t * 16 + lane15;
    float bv = bias ? bias[col] : 0.f;
#pragma unroll
    for (int r = 0; r < 8; ++r) {
      int row0 = rowBase + hi * 8 + r;
      if (row0 < nRows) {
        float v = acc0[t][r] + bv;
        if (flags & 1) v = fmaxf(v, 0.f);
        size_t o = (size_t)row0 * Ncols + col;
        if (flags & 2) v += out[o];
        out[o] = v;
      }
      int row1 = rowBase + 16 + hi * 8 + r;
      if (row1 < nRows) {
        float v = acc1[t][r] + bv;
        if (flags & 1) v = fmaxf(v, 0.f);
        size_t o = (size_t)row1 * Ncols + col;
        if (flags & 2) v += out[o];
        out[o] = v;
      }
    }
  }
}

// ---------------- GCN ----------------
__global__ __launch_bounds__(256) void deg_count(const int* __restrict__ dst, int E,
                                                 float* __restrict__ deg) {
  int i = blockIdx.x * 256 + threadIdx.x;
  if (i < E) atomicAdd(&deg[dst[i]], 1.0f);
}
__global__ __launch_bounds__(256) void dinv_k(float* __restrict__ d, int n) {
  int i = blockIdx.x * 256 + threadIdx.x;
  if (i < n) d[i] = rsqrtf(d[i]);   // deg >= 1 (self loop)
}
// one wave per edge, 4 channels per lane
__global__ __launch_bounds__(256) void gcn_scatter(
    const int* __restrict__ src, const int* __restrict__ dst, int E,
    const float* __restrict__ xw, const float* __restrict__ dinv, float* __restrict__ acc) {
  int i = blockIdx.x * 256 + threadIdx.x;
  int e = i >> 5;
  if (e >= E) return;
  int q = (i & 31) * 4;
  int s = src[e], t = dst[e];
  float nrm = dinv[s] * dinv[t];
  float4 v = *(const float4*)(xw + (size_t)s * DIM + q);
  float* a = acc + (size_t)t * DIM + q;
  atomicAdd(a + 0, nrm * v.x);
  atomicAdd(a + 1, nrm * v.y);
  atomicAdd(a + 2, nrm * v.z);
  atomicAdd(a + 3, nrm * v.w);
}
// acc = x + acc + dinv^2 * xw + gcn_b   (residual + self loop + bias, pre-BN)
__global__ __launch_bounds__(256) void gcn_finalize(
    const float* __restrict__ x, const float* __restrict__ xw,
    const float* __restrict__ dinv, const float* __restrict__ gb,
    float* __restrict__ acc, int n) {
  int i = blockIdx.x * 256 + threadIdx.x;
  if (i >= n * DIM) return;
  int row = i >> 7, col = i & 127;
  float dv = dinv[row];
  acc[i] = x[i] + acc[i] + dv * dv * xw[i] + gb[col];
}

// ---------------- BatchNorm (per-column over rows) ----------------
__global__ __launch_bounds__(DIM) void bn_stats(const float* __restrict__ x, int n,
                                                float* __restrict__ stats) {
  int col = threadIdx.x;
  float s = 0.f, s2 = 0.f;
  for (int r = blockIdx.x; r < n; r += gridDim.x) {
    float v = x[(size_t)r * DIM + col];
    s += v; s2 += v * v;
  }
  atomicAdd(&stats[col], s);
  atomicAdd(&stats[DIM + col], s2);
}
__global__ __launch_bounds__(256) void bn_apply(
    const float* __restrict__ x, const float* __restrict__ g, const float* __restrict__ b,
    const float* __restrict__ stats, float invN, float* __restrict__ out, int n) {
  int i = blockIdx.x * 256 + threadIdx.x;
  if (i >= n * DIM) return;
  int col = i & 127;
  float mu  = stats[col] * invN;
  float var = stats[DIM + col] * invN - mu * mu;
  out[i] = g[col] * (x[i] - mu) * rsqrtf(var + 1e-5f) + b[col];
}

// ---------------- GATv2 edge kernels ----------------
__device__ __forceinline__ unsigned enc_f(float f) {
  unsigned u = __float_as_uint(f);
  return (u & 0x80000000u) ? ~u : (u | 0x80000000u);
}
__device__ __forceinline__ float dec_f(unsigned e) {
  unsigned b = (e & 0x80000000u) ? (e ^ 0x80000000u) : ~e;
  return __uint_as_float(b);
}
// thread per (edge, head)
__global__ __launch_bounds__(256) void gat_logits(
    const int* __restrict__ src, const int* __restrict__ dst, int E,
    const float* __restrict__ xs, const float* __restrict__ xd,
    const float* __restrict__ att, float* __restrict__ logits,
    unsigned* __restrict__ maxenc) {
  int i = blockIdx.x * 256 + threadIdx.x;
  int e = i >> 3;
  if (e >= E) return;
  int h = i & 7;
  int s = src[e], t = dst[e];
  const float* pj = xs + (size_t)s * DIM + h * CCH;
  const float* pi = xd + (size_t)t * DIM + h * CCH;
  const float* pa = att + h * CCH;
  float sum = 0.f;
#pragma unroll
  for (int c = 0; c < CCH; ++c) {
    float z = pi[c] + pj[c];
    z = z > 0.f ? z : 0.2f * z;        // leaky_relu, slope 0.2
    sum += z * pa[c];
  }
  logits[(size_t)e * 8 + h] = sum;
  atomicMax(&maxenc[(size_t)t * 8 + h], enc_f(sum));
}
__global__ __launch_bounds__(256) void gat_expsum(
    const int* __restrict__ dst, int E, float* __restrict__ logits,
    const unsigned* __restrict__ maxenc, float* __restrict__ sumexp) {
  int i = blockIdx.x * 256 + threadIdx.x;
  int e = i >> 3;
  if (e >= E) return;
  int h = i & 7;
  int t = dst[e];
  float a = __expf(logits[i] - dec_f(maxenc[(size_t)t * 8 + h]));
  logits[i] = a;
  atomicAdd(&sumexp[(size_t)t * 8 + h], a);
}
// one wave per edge, 4 channels per lane; head = lane/4
__global__ __launch_bounds__(256) void gat_scatter(
    const int* __restrict__ src, const int* __restrict__ dst, int E,
    const float* __restrict__ xs, const float* __restrict__ logits,
    const float* __restrict__ sumexp, float* __restrict__ acc) {
  int i = blockIdx.x * 256 + threadIdx.x;
  int e = i >> 5;
  if (e >= E) return;
  int lane = i & 31, q = lane * 4, h = lane >> 2;
  int s = src[e], t = dst[e];
  float alpha = logits[(size_t)e * 8 + h] / sumexp[(size_t)t * 8 + h];
  float4 v = *(const float4*)(xs + (size_t)s * DIM + q);
  float* a = acc + (size_t)t * DIM + q;
  atomicAdd(a + 0, alpha * v.x);
  atomicAdd(a + 1, alpha * v.y);
  atomicAdd(a + 2, alpha * v.z);
  atomicAdd(a + 3, alpha * v.w);
}

// ---------------- elementwise glue ----------------
__global__ __launch_bounds__(256) void add_bias_ip(float* __restrict__ p,
                                                   const float* __restrict__ bias, int n) {
  int i = blockIdx.x * 256 + threadIdx.x;
  if (i < n * DIM) p[i] += bias[i & 127];
}
__global__ __launch_bounds__(256) void residual_bias(
    const float* __restrict__ x, const float* __restrict__ bias, float* __restrict__ p, int n) {
  int i = blockIdx.x * 256 + threadIdx.x;
  if (i < n * DIM) p[i] = x[i] + p[i] + bias[i & 127];
}
// o1 = o2 = a + b  (h stored in two buffers: one stays read-only for GEMM input)
__global__ __launch_bounds__(256) void add_dual(
    const float* __restrict__ a, const float* __restrict__ b,
    float* __restrict__ o1, float* __restrict__ o2, int cnt) {
  int i = blockIdx.x * 256 + threadIdx.x;
  if (i < cnt) { float v = a[i] + b[i]; o1[i] = v; o2[i] = v; }
}

// ---------------- host-side launch ----------------
static inline unsigned nblk(long long n) { return (unsigned)((n + 255) / 256); }

extern "C" void kernel_launch(void* const* d_in, const int* in_sizes, int n_in,
                              void* d_out, int out_size, void* d_ws, size_t ws_size,
                              hipStream_t stream) {
  const float* x    = (const float*)d_in[0];
  const float* hubf = (const float*)d_in[1];
  const int*   ei   = (const int*)d_in[2];
  const int*   sh   = (const int*)d_in[3];
  const int*   hh   = (const int*)d_in[4];
  const float* gcn_W = (const float*)d_in[5];
  const float* gcn_b = (const float*)d_in[6];
  const float *s2h_Wl = (const float*)d_in[7],  *s2h_bl = (const float*)d_in[8];
  const float *s2h_Wr = (const float*)d_in[9],  *s2h_br = (const float*)d_in[10];
  const float *s2h_att = (const float*)d_in[11], *s2h_b = (const float*)d_in[12];
  const float *hh_Wl = (const float*)d_in[13], *hh_bl = (const float*)d_in[14];
  const float *hh_Wr = (const float*)d_in[15], *hh_br = (const float*)d_in[16];
  const float *hh_att = (const float*)d_in[17], *hh_b = (const float*)d_in[18];
  const float *h2s_Wl = (const float*)d_in[19], *h2s_bl = (const float*)d_in[20];
  const float *h2s_Wr = (const float*)d_in[21], *h2s_br = (const float*)d_in[22];
  const float *h2s_att = (const float*)d_in[23], *h2s_b = (const float*)d_in[24];
  const float *bn1l_g = (const float*)d_in[25], *bn1l_b = (const float*)d_in[26];
  const float *bn1a_g = (const float*)d_in[27], *bn1a_b = (const float*)d_in[28];
  const float *bn2_g  = (const float*)d_in[29], *bn2_b  = (const float*)d_in[30];
  const float *ff_W1 = (const float*)d_in[31], *ff_b1 = (const float*)d_in[32];
  const float *ff_W2 = (const float*)d_in[33], *ff_b2 = (const float*)d_in[34];

  const int N   = in_sizes[0] / DIM;   // 50000
  const int M   = in_sizes[1] / DIM;   // 500
  const int E   = in_sizes[2] / 2;     // 800000
  const int Esh = in_sizes[3] / 2;     // 150000
  const int Ehh = in_sizes[4] / 2;     // 5000

  const int* e_src = ei;         const int* e_dst = ei + E;
  const int* spoke = sh;         const int* hubid = sh + Esh;
  const int* hh_src = hh;        const int* hh_dst = hh + Ehh;

  // workspace layout (floats)
  float* ws = (float*)d_ws;
  const size_t NB = (size_t)N * DIM;
  float* WS0  = ws;                         // [N*128]
  float* WS1  = WS0 + NB;                   // [N*128]
  float* WS2  = WS1 + NB;                   // [N*128] (h_local)
  float* WSF  = WS2 + NB;                   // [N*256] FFN mid
  float* WSL  = WSF + (size_t)N * 2 * DIM;  // [Esh*8] logits/alpha
  float* WSMf = WSL + (size_t)Esh * 8;      // [N*8] deg / maxenc
  unsigned* WSMu = (unsigned*)WSMf;
  float* WSS  = WSMf + (size_t)N * 8;       // [N*8] sumexp
  float* WSH0 = WSS + (size_t)N * 8;        // [M*128]
  float* WSH1 = WSH0 + (size_t)M * DIM;     // [M*128]
  float* WSH2 = WSH1 + (size_t)M * DIM;     // [M*128]
  float* WSH3 = WSH2 + (size_t)M * DIM;     // [M*128]
  float* STATS = WSH3 + (size_t)M * DIM;    // [256]

  const unsigned ENC_NEGINF = 0x007FFFFFu;  // enc(-inf)
  const float invN = 1.0f / (float)N;

  dim3 g128((N + 255) / 256, 2);      // Ncols=128
  dim3 g256((N + 255) / 256, 4);      // Ncols=256
  dim3 gHub((M + 255) / 256, 2);

  // ---- GCN: h_localpre = x + segsum(norm*xw) + selfloop + b ; BN -> h_local (WS2)
  fill_f32<<<nblk(N), 256, 0, stream>>>(WSMf, N, 1.0f);                 // deg (incl. self loop)
  deg_count<<<nblk(E), 256, 0, stream>>>(e_dst, E, WSMf);
  dinv_k<<<nblk(N), 256, 0, stream>>>(WSMf, N);                         // deg -> dinv
  gemm_wmma<<<g128, 256, 0, stream>>>(x, gcn_W, nullptr, WS0, N, DIM, DIM, 0);   // xw
  fill_f32<<<nblk(NB), 256, 0, stream>>>(WS1, (int)NB, 0.f);
  gcn_scatter<<<nblk((long long)E * 32), 256, 0, stream>>>(e_src, e_dst, E, WS0, WSMf, WS1);
  gcn_finalize<<<nblk(NB), 256, 0, stream>>>(x, WS0, WSMf, gcn_b, WS1, N);
  fill_f32<<<1, 256, 0, stream>>>(STATS, 256, 0.f);
  bn_stats<<<512, DIM, 0, stream>>>(WS1, N, STATS);
  bn_apply<<<nblk(NB), 256, 0, stream>>>(WS1, bn1l_g, bn1l_b, STATS, invN, WS2, N);

  // ---- GATv2 spokes -> hubs : hg = WSH1
  gemm_wmma<<<g128, 256, 0, stream>>>(WS2, s2h_Wl, s2h_bl, WS0, N, DIM, DIM, 0);   // xj (spokes)
  gemm_wmma<<<gHub, 256, 0, stream>>>(hubf, s2h_Wr, s2h_br, WSH0, M, DIM, DIM, 0); // xi (hubs)
  fill_u32<<<nblk(M * 8), 256, 0, stream>>>(WSMu, M * 8, ENC_NEGINF);
  fill_f32<<<nblk(M * 8), 256, 0, stream>>>(WSS, M * 8, 0.f);
  gat_logits<<<nblk((long long)Esh * 8), 256, 0, stream>>>(spoke, hubid, Esh, WS0, WSH0, s2h_att, WSL, WSMu);
  gat_expsum<<<nblk((long long)Esh * 8), 256, 0, stream>>>(hubid, Esh, WSL, WSMu, WSS);
  fill_f32<<<nblk((long long)M * DIM), 256, 0, stream>>>(WSH1, M * DIM, 0.f);
  gat_scatter<<<nblk((long long)Esh * 32), 256, 0, stream>>>(spoke, hubid, Esh, WS0, WSL, WSS, WSH1);
  add_bias_ip<<<nblk((long long)M * DIM), 256, 0, stream>>>(WSH1, s2h_b, M);

  // ---- GATv2 hubs <-> hubs : hg2 = WSH0
  gemm_wmma<<<gHub, 256, 0, stream>>>(WSH1, hh_Wl, hh_bl, WSH2, M, DIM, DIM, 0);
  gemm_wmma<<<gHub, 256, 0, stream>>>(WSH1, hh_Wr, hh_br, WSH3, M, DIM, DIM, 0);
  fill_u32<<<nblk(M * 8), 256, 0, stream>>>(WSMu, M * 8, ENC_NEGINF);
  fill_f32<<<nblk(M * 8), 256, 0, stream>>>(WSS, M * 8, 0.f);
  gat_logits<<<nblk((long long)Ehh * 8), 256, 0, stream>>>(hh_src, hh_dst, Ehh, WSH2, WSH3, hh_att, WSL, WSMu);
  gat_expsum<<<nblk((long long)Ehh * 8), 256, 0, stream>>>(hh_dst, Ehh, WSL, WSMu, WSS);
  fill_f32<<<nblk((long long)M * DIM), 256, 0, stream>>>(WSH0, M * DIM, 0.f);
  gat_scatter<<<nblk((long long)Ehh * 32), 256, 0, stream>>>(hh_src, hh_dst, Ehh, WSH2, WSL, WSS, WSH0);
  add_bias_ip<<<nblk((long long)M * DIM), 256, 0, stream>>>(WSH0, hh_b, M);

  // ---- GATv2 hubs -> spokes : h_attn acc in WS1
  gemm_wmma<<<gHub, 256, 0, stream>>>(WSH0, h2s_Wl, h2s_bl, WSH2, M, DIM, DIM, 0); // xj (hubs)
  gemm_wmma<<<g128, 256, 0, stream>>>(WS2, h2s_Wr, h2s_br, WS0, N, DIM, DIM, 0);   // xi (spokes)
  fill_u32<<<nblk((long long)N * 8), 256, 0, stream>>>(WSMu, N * 8, ENC_NEGINF);
  fill_f32<<<nblk((long long)N * 8), 256, 0, stream>>>(WSS, N * 8, 0.f);
  gat_logits<<<nblk((long long)Esh * 8), 256, 0, stream>>>(hubid, spoke, Esh, WSH2, WS0, h2s_att, WSL, WSMu);
  gat_expsum<<<nblk((long long)Esh * 8), 256, 0, stream>>>(spoke, Esh, WSL, WSMu, WSS);
  fill_f32<<<nblk(NB), 256, 0, stream>>>(WS1, (int)NB, 0.f);
  gat_scatter<<<nblk((long long)Esh * 32), 256, 0, stream>>>(hubid, spoke, Esh, WSH2, WSL, WSS, WS1);
  residual_bias<<<nblk(NB), 256, 0, stream>>>(x, h2s_b, WS1, N);       // h_in + h_attn + b
  fill_f32<<<1, 256, 0, stream>>>(STATS, 256, 0.f);
  bn_stats<<<512, DIM, 0, stream>>>(WS1, N, STATS);
  bn_apply<<<nblk(NB), 256, 0, stream>>>(WS1, bn1a_g, bn1a_b, STATS, invN, WS1, N); // h_attn (in place)

  // ---- combine + FFN + final BN -> d_out
  add_dual<<<nblk(NB), 256, 0, stream>>>(WS2, WS1, WS0, WS1, (int)NB); // h in WS0 (read) and WS1 (accum)
  gemm_wmma<<<g256, 256, 0, stream>>>(WS0, ff_W1, ff_b1, WSF, N, DIM, 2 * DIM, 1);       // relu
  gemm_wmma<<<g128, 256, 0, stream>>>(WSF, ff_W2, ff_b2, WS1, N, 2 * DIM, DIM, 2);       // accumulate
  fill_f32<<<1, 256, 0, stream>>>(STATS, 256, 0.f);
  bn_stats<<<512, DIM, 0, stream>>>(WS1, N, STATS);
  bn_apply<<<nblk(NB), 256, 0, stream>>>(WS1, bn2_g, bn2_b, STATS, invN, (float*)d_out, N);
}